// EGATModel_26482768347805
// MI455X (gfx1250) — compile-verified
//
#include <hip/hip_runtime.h>
#include <hip/hip_bf16.h>
#include <cstddef>
#include <cstdint>

// ---------------- model constants ----------------
#define HEADS 4
#define NEG_SLOPE 0.01f
#define BN_EPS 1e-5f
#define BATCH 8
#define NPG 2048
#define EPG 16384
#define NNODES (BATCH * NPG)     // 16384
#define NEDGES (BATCH * EPG)     // 131072
#define D_IN 64
#define NHID 128
#define S2S_ITERS 3

typedef __attribute__((ext_vector_type(2))) float v2f;
typedef __attribute__((ext_vector_type(8))) float v8f;

// ---------------- small helpers ----------------
__device__ __forceinline__ float sigmoidf(float x) { return 1.0f / (1.0f + expf(-x)); }

__device__ __forceinline__ void atomicMaxF(float* addr, float val) {
    unsigned int* u = (unsigned int*)addr;
    unsigned int old = *u;
    while (__uint_as_float(old) < val) {
        unsigned int assumed = old;
        old = atomicCAS(u, assumed, __float_as_uint(val));
        if (old == assumed) break;
    }
}

__global__ void k_fill(float* __restrict__ p, size_t n, float v) {
    size_t i = (size_t)blockIdx.x * blockDim.x + threadIdx.x;
    if (i < n) p[i] = v;
}

// ---------------- WMMA GEMM: C(M,Nc) = A(M,K) @ W(K,Nc), all fp32 row-major ----------
// grid.x = M/16, grid.y = Nc/64, block = 32 (one wave). Uses V_WMMA_F32_16X16X4_F32.
__global__ void k_gemm_rm(const float* __restrict__ A, const float* __restrict__ W,
                          float* __restrict__ C, int M, int K, int Nc) {
    const int lane = threadIdx.x;
    const int half = lane >> 4;
    const int l    = lane & 15;
    const int m0   = blockIdx.x * 16;
    const int n0   = blockIdx.y * 64;

    v8f acc0 = {}, acc1 = {}, acc2 = {}, acc3 = {};
    const float* arow = A + (size_t)(m0 + l) * K + 2 * half;
    for (int k = 0; k < K; k += 4) {
        v2f a;
        a.x = arow[k];
        a.y = arow[k + 1];
        __builtin_prefetch(arow + k + 4, 0, 0);
        const float* bbase = W + (size_t)(k + 2 * half) * Nc + n0 + l;
        v2f b0; b0.x = bbase[0];  b0.y = bbase[Nc];
        v2f b1; b1.x = bbase[16]; b1.y = bbase[Nc + 16];
        v2f b2; b2.x = bbase[32]; b2.y = bbase[Nc + 32];
        v2f b3; b3.x = bbase[48]; b3.y = bbase[Nc + 48];
        acc0 = __builtin_amdgcn_wmma_f32_16x16x4_f32(false, a, false, b0, (short)0, acc0, false, false);
        acc1 = __builtin_amdgcn_wmma_f32_16x16x4_f32(false, a, false, b1, (short)0, acc1, false, false);
        acc2 = __builtin_amdgcn_wmma_f32_16x16x4_f32(false, a, false, b2, (short)0, acc2, false, false);
        acc3 = __builtin_amdgcn_wmma_f32_16x16x4_f32(false, a, false, b3, (short)0, acc3, false, false);
    }
#pragma unroll
    for (int r = 0; r < 8; ++r) {
        float* crow = C + (size_t)(m0 + r + 8 * half) * Nc + n0 + l;
        crow[0]  = acc0[r];
        crow[16] = acc1[r];
        crow[32] = acc2[r];
        crow[48] = acc3[r];
    }
}

// ---------------- fused edge kernel --------------------------------------------------
// For a block of 16 edges: F = ef_tile(16,Din) @ Wf(Din,Kout) via WMMA; epilogue adds
// Pni[src], Pnj[dst], bias; leaky-relu; accumulates attention logits e (16,H) and
// head-mean ef_next (16,De) in LDS; writes both. Never materializes (E,Kout).
__global__ void k_edge_fused(const float* __restrict__ ef, int Din,
                             const float* __restrict__ Wf, int Kout, int De,
                             const float* __restrict__ Pni, const float* __restrict__ Pnj,
                             const float* __restrict__ bias, const float* __restrict__ attn,
                             const int* __restrict__ src, const int* __restrict__ dst,
                             float* __restrict__ e_out, float* __restrict__ ef_next) {
    __shared__ float As[16 * 129];      // A tile, padded stride (Din+1) <= 129
    __shared__ float e_s[16 * HEADS];   // attention logit accumulators
    __shared__ float ef_acc[16 * 128];  // head-mean accumulators (stride 128)

    const int tid  = threadIdx.x;       // 0..31
    const int half = tid >> 4;
    const int l    = tid & 15;
    const int e0   = blockIdx.x * 16;
    const int strideA = Din + 1;

    for (int i = tid; i < 16 * HEADS; i += 32) e_s[i] = 0.0f;
    for (int i = tid; i < 16 * 128; i += 32) ef_acc[i] = 0.0f;
    for (int i = tid; i < 16 * Din; i += 32) {
        int row = i / Din, col = i % Din;
        As[row * strideA + col] = ef[(size_t)(e0 + row) * Din + col];
    }
    __syncthreads();

    for (int n0 = 0; n0 < Kout; n0 += 16) {
        v8f acc = {};
        for (int k = 0; k < Din; k += 4) {
            v2f a;
            a.x = As[l * strideA + k + 2 * half];
            a.y = As[l * strideA + k + 2 * half + 1];
            const float* bbase = Wf + (size_t)(k + 2 * half) * Kout + n0 + l;
            v2f b; b.x = bbase[0]; b.y = bbase[Kout];
            acc = __builtin_amdgcn_wmma_f32_16x16x4_f32(false, a, false, b, (short)0, acc, false, false);
        }
        const int h  = n0 / De;
        const int d0 = n0 - h * De;
        const float attn_v = attn[h * De + d0 + l];
        const float bias_v = bias[n0 + l];
#pragma unroll
        for (int r = 0; r < 8; ++r) {
            const int row  = r + 8 * half;
            const int eidx = e0 + row;
            const int sn = src[eidx];
            const int dn = dst[eidx];
            float v = acc[r] + Pni[(size_t)sn * Kout + n0 + l]
                             + Pnj[(size_t)dn * Kout + n0 + l] + bias_v;
            v = v > 0.0f ? v : NEG_SLOPE * v;
            atomicAdd(&e_s[row * HEADS + h], v * attn_v);
            atomicAdd(&ef_acc[row * 128 + d0 + l], v * 0.25f);  // mean over 4 heads
        }
    }
    __syncthreads();
    for (int i = tid; i < 16 * HEADS; i += 32)
        e_out[(size_t)(e0 + i / HEADS) * HEADS + (i % HEADS)] = e_s[i];
    for (int i = tid; i < 16 * De; i += 32)
        ef_next[(size_t)(e0 + i / De) * De + (i % De)] = ef_acc[(i / De) * 128 + (i % De)];
}

// ---------------- segmented softmax over dst -----------------------------------------
__global__ void k_seg_max(const float* __restrict__ e, const int* __restrict__ dst,
                          float* __restrict__ node_m) {
    size_t i = (size_t)blockIdx.x * blockDim.x + threadIdx.x;  // over E*H
    int edge = (int)(i / HEADS), h = (int)(i % HEADS);
    atomicMaxF(&node_m[(size_t)dst[edge] * HEADS + h], e[i]);
}

__global__ void k_exp_sum(float* __restrict__ e, const int* __restrict__ dst,
                          const float* __restrict__ node_m, float* __restrict__ node_s) {
    size_t i = (size_t)blockIdx.x * blockDim.x + threadIdx.x;  // over E*H
    int edge = (int)(i / HEADS), h = (int)(i % HEADS);
    float m = node_m[(size_t)dst[edge] * HEADS + h];
    float ex = expf(e[i] - m);
    e[i] = ex;
    atomicAdd(&node_s[(size_t)dst[edge] * HEADS + h], ex);
}

// ---------------- aggregation: h_out[dst] += Pnode[src] * a --------------------------
__global__ void k_aggregate(const float* __restrict__ ex, const float* __restrict__ node_s,
                            const int* __restrict__ src, const int* __restrict__ dst,
                            const float* __restrict__ Pnode, float* __restrict__ h_out,
                            int Kout, int Dn) {
    size_t i = (size_t)blockIdx.x * blockDim.x + threadIdx.x;  // over E*Kout
    int edge = (int)(i / Kout), c = (int)(i % Kout);
    int h = c / Dn;
    int dn = dst[edge];
    float a = ex[(size_t)edge * HEADS + h] / node_s[(size_t)dn * HEADS + h];
    atomicAdd(&h_out[(size_t)dn * Kout + c], Pnode[(size_t)src[edge] * Kout + c] * a);
}

// ---------------- head mean (+ optional residual) ------------------------------------
__global__ void k_headmean(const float* __restrict__ h_out, const float* __restrict__ resid,
                           float* __restrict__ nf_out, int Dn) {
    size_t i = (size_t)blockIdx.x * blockDim.x + threadIdx.x;  // over N*Dn
    int n = (int)(i / Dn), d = (int)(i % Dn);
    const float* row = h_out + (size_t)n * (HEADS * Dn);
    float s = 0.25f * (row[d] + row[Dn + d] + row[2 * Dn + d] + row[3 * Dn + d]);
    if (resid) s += resid[(size_t)n * Dn + d];
    nf_out[i] = s;
}

// ---------------- batchnorm ----------------------------------------------------------
__global__ void k_bn_stats(const float* __restrict__ x, int Nrow, int C,
                           float* __restrict__ mu, float* __restrict__ var) {
    int c = blockIdx.x;
    float s = 0.0f, s2 = 0.0f;
    for (int n = threadIdx.x; n < Nrow; n += blockDim.x) {
        float v = x[(size_t)n * C + c];
        s += v; s2 += v * v;
    }
    __shared__ float rs[256], rs2[256];
    rs[threadIdx.x] = s; rs2[threadIdx.x] = s2;
    __syncthreads();
    for (int off = 128; off > 0; off >>= 1) {
        if (threadIdx.x < off) { rs[threadIdx.x] += rs[threadIdx.x + off]; rs2[threadIdx.x] += rs2[threadIdx.x + off]; }
        __syncthreads();
    }
    if (threadIdx.x == 0) {
        float m = rs[0] / (float)Nrow;
        mu[c] = m;
        var[c] = rs2[0] / (float)Nrow - m * m;
    }
}

__global__ void k_bn_apply(float* __restrict__ x, const float* __restrict__ mu,
                           const float* __restrict__ var, const float* __restrict__ gamma,
                           const float* __restrict__ beta, int C) {
    size_t i = (size_t)blockIdx.x * blockDim.x + threadIdx.x;  // over N*C
    int c = (int)(i % C);
    x[i] = (x[i] - mu[c]) * rsqrtf(var[c] + BN_EPS) * gamma[c] + beta[c];
}

// ---------------- set2set LSTM step --------------------------------------------------
// grid 2 x 256; thread t -> (b = t/64, d = t%64). x = q_star (B,128).
__global__ void k_lstm(const float* __restrict__ q, float* __restrict__ h, float* __restrict__ c,
                       const float* __restrict__ Wih, const float* __restrict__ Whh,
                       const float* __restrict__ bih, const float* __restrict__ bhh) {
    __shared__ float hsh[256];  // old h for this block's 4 graphs
    int tid = threadIdx.x;
    int t = blockIdx.x * 256 + tid;
    hsh[tid] = h[t];            // copy 4 graphs worth of h
    __syncthreads();
    int b = t / 64, d = t % 64;
    int bb = tid / 64;
    const float* qb = q + (size_t)b * 128;
    float zi = bih[d]       + bhh[d];
    float zf = bih[64 + d]  + bhh[64 + d];
    float zg = bih[128 + d] + bhh[128 + d];
    float zo = bih[192 + d] + bhh[192 + d];
    for (int k = 0; k < 128; ++k) {
        float x = qb[k];
        zi += x * Wih[(size_t)d * 128 + k];
        zf += x * Wih[(size_t)(64 + d) * 128 + k];
        zg += x * Wih[(size_t)(128 + d) * 128 + k];
        zo += x * Wih[(size_t)(192 + d) * 128 + k];
    }
    for (int k = 0; k < 64; ++k) {
        float x = hsh[bb * 64 + k];
        zi += x * Whh[(size_t)d * 64 + k];
        zf += x * Whh[(size_t)(64 + d) * 64 + k];
        zg += x * Whh[(size_t)(128 + d) * 64 + k];
        zo += x * Whh[(size_t)(192 + d) * 64 + k];
    }
    __syncthreads();
    float cc = sigmoidf(zf) * c[t] + sigmoidf(zi) * tanhf(zg);
    c[t] = cc;
    h[t] = sigmoidf(zo) * tanhf(cc);
}

// ---------------- set2set attention (one block per graph; nodes contiguous) ----------
__global__ void k_s2s_attn(const float* __restrict__ feat, const float* __restrict__ h,
                           float* __restrict__ q_star) {
    __shared__ float el[NPG];
    __shared__ float red[256];
    __shared__ float hs[64];
    int b = blockIdx.x, tid = threadIdx.x;
    if (tid < 64) hs[tid] = h[(size_t)b * 64 + tid];
    __syncthreads();
    float lmax = -3.0e38f;
    for (int n = tid; n < NPG; n += 256) {
        const float* f = feat + (size_t)(b * NPG + n) * 64;
        float d = 0.0f;
        for (int k = 0; k < 64; ++k) d += f[k] * hs[k];
        el[n] = d;
        lmax = fmaxf(lmax, d);
    }
    red[tid] = lmax; __syncthreads();
    for (int off = 128; off > 0; off >>= 1) {
        if (tid < off) red[tid] = fmaxf(red[tid], red[tid + off]);
        __syncthreads();
    }
    float m = red[0]; __syncthreads();
    float ls = 0.0f;
    for (int n = tid; n < NPG; n += 256) {
        float ex = expf(el[n] - m);
        el[n] = ex;
        ls += ex;
    }
    red[tid] = ls; __syncthreads();
    for (int off = 128; off > 0; off >>= 1) {
        if (tid < off) red[tid] += red[tid + off];
        __syncthreads();
    }
    float ssum = red[0]; __syncthreads();
    for (int d0 = 0; d0 < 64; ++d0) {
        float lr = 0.0f;
        for (int n = tid; n < NPG; n += 256)
            lr += el[n] * feat[(size_t)(b * NPG + n) * 64 + d0];
        red[tid] = lr; __syncthreads();
        for (int off = 128; off > 0; off >>= 1) {
            if (tid < off) red[tid] += red[tid + off];
            __syncthreads();
        }
        if (tid == 0) q_star[(size_t)b * 128 + 64 + d0] = red[0] / ssum;
        __syncthreads();
    }
    if (tid < 64) q_star[(size_t)b * 128 + tid] = hs[tid];
}

// ---------------- final MLP head -----------------------------------------------------
__global__ void k_head_mlp(const float* __restrict__ qd, const float* __restrict__ qa,
                           const float* __restrict__ w1, const float* __restrict__ b1,
                           const float* __restrict__ w2, const float* __restrict__ b2,
                           const float* __restrict__ w3, const float* __restrict__ b3,
                           float* __restrict__ out) {
    __shared__ float x[BATCH * 256];
    __shared__ float y1[BATCH * 128];
    __shared__ float y2[BATCH * 32];
    int tid = threadIdx.x;
    for (int i = tid; i < BATCH * 128; i += 256) {
        int b = i / 128, d = i % 128;
        x[b * 256 + d] = qd[i];
        x[b * 256 + 128 + d] = qa[i];
    }
    __syncthreads();
    for (int i = tid; i < BATCH * 128; i += 256) {
        int b = i / 128, j = i % 128;
        float s = b1[j];
        for (int k = 0; k < 256; ++k) s += x[b * 256 + k] * w1[k * 128 + j];
        y1[i] = fmaxf(s, 0.0f);
    }
    __syncthreads();
    for (int i = tid; i < BATCH * 32; i += 256) {
        int b = i / 32, j = i % 32;
        float s = b2[j];
        for (int k = 0; k < 128; ++k) s += y1[b * 128 + k] * w2[k * 32 + j];
        y2[i] = fmaxf(s, 0.0f);
    }
    __syncthreads();
    for (int i = tid; i < BATCH * 2; i += 256) {
        int b = i / 2, j = i % 2;
        float s = b3[j];
        for (int k = 0; k < 32; ++k) s += y2[b * 32 + k] * w3[k * 2 + j];
        out[i] = s;
    }
}

// ---------------- host orchestration -------------------------------------------------
struct WS {
    float *Pni, *Pnj, *Pnode, *e_att, *node_m, *node_s, *h_out;
    float *nfA, *nfB, *efA, *efB, *bn_mu, *bn_var;
    float *s2s_h, *s2s_c, *q_don, *q_acc;
};

static inline void fill_f(float* p, size_t n, float v, hipStream_t s) {
    int blocks = (int)((n + 255) / 256);
    k_fill<<<blocks, 256, 0, s>>>(p, n, v);
}

static void run_tower(const float* nf0, const float* ef0, const int* src, const int* dst,
                      void* const* P, WS& w, float* q_star, hipStream_t s) {
    const int Din[3]  = {D_IN, NHID, D_IN};       // 64, 128, 64
    const int Kout[3] = {HEADS * NHID, HEADS * (NHID / 2), HEADS * D_IN};  // 512, 256, 256

    const float* nf_in = nf0;
    const float* ef_in = ef0;
    float* nf_bufs[2] = {w.nfA, w.nfB};
    float* ef_bufs[2] = {w.efA, w.efB};

    for (int l = 0; l < 3; ++l) {
        const int din = Din[l], kout = Kout[l], dn = kout / HEADS;
        const float* Wn   = (const float*)P[6 * l + 0];
        const float* Wni  = (const float*)P[6 * l + 1];
        const float* Wnj  = (const float*)P[6 * l + 2];
        const float* Wf   = (const float*)P[6 * l + 3];
        const float* attn = (const float*)P[6 * l + 4];
        const float* bias = (const float*)P[6 * l + 5];

        fill_f(w.node_m, (size_t)NNODES * HEADS, -3.0e38f, s);
        fill_f(w.node_s, (size_t)NNODES * HEADS, 0.0f, s);
        fill_f(w.h_out, (size_t)NNODES * kout, 0.0f, s);

        dim3 gg(NNODES / 16, kout / 64);
        k_gemm_rm<<<gg, 32, 0, s>>>(nf_in, Wni, w.Pni, NNODES, din, kout);
        k_gemm_rm<<<gg, 32, 0, s>>>(nf_in, Wnj, w.Pnj, NNODES, din, kout);
        k_gemm_rm<<<gg, 32, 0, s>>>(nf_in, Wn,  w.Pnode, NNODES, din, kout);

        float* ef_out = ef_bufs[l & 1];
        k_edge_fused<<<NEDGES / 16, 32, 0, s>>>(ef_in, din, Wf, kout, dn,
                                                w.Pni, w.Pnj, bias, attn, src, dst,
                                                w.e_att, ef_out);

        k_seg_max<<<(NEDGES * HEADS) / 256, 256, 0, s>>>(w.e_att, dst, w.node_m);
        k_exp_sum<<<(NEDGES * HEADS) / 256, 256, 0, s>>>(w.e_att, dst, w.node_m, w.node_s);

        size_t agg = (size_t)NEDGES * kout;
        k_aggregate<<<(int)(agg / 256), 256, 0, s>>>(w.e_att, w.node_s, src, dst,
                                                     w.Pnode, w.h_out, kout, dn);

        float* nf_out = nf_bufs[l & 1];
        size_t nm = (size_t)NNODES * dn;
        k_headmean<<<(int)(nm / 256), 256, 0, s>>>(w.h_out, (l == 2) ? nf0 : nullptr, nf_out, dn);

        if (l < 2) {
            const float* gamma = (const float*)P[18 + l];
            const float* beta  = (const float*)P[20 + l];
            k_bn_stats<<<dn, 256, 0, s>>>(nf_out, NNODES, dn, w.bn_mu, w.bn_var);
            k_bn_apply<<<(int)(nm / 256), 256, 0, s>>>(nf_out, w.bn_mu, w.bn_var, gamma, beta, dn);
        }
        nf_in = nf_out;
        ef_in = ef_out;
    }

    // set2set on nf_in (N x 64)
    const float* Wih = (const float*)P[22];
    const float* Whh = (const float*)P[23];
    const float* bih = (const float*)P[24];
    const float* bhh = (const float*)P[25];
    fill_f(q_star, BATCH * 128, 0.0f, s);
    fill_f(w.s2s_h, BATCH * 64, 0.0f, s);
    fill_f(w.s2s_c, BATCH * 64, 0.0f, s);
    for (int it = 0; it < S2S_ITERS; ++it) {
        k_lstm<<<2, 256, 0, s>>>(q_star, w.s2s_h, w.s2s_c, Wih, Whh, bih, bhh);
        k_s2s_attn<<<BATCH, 256, 0, s>>>(nf_in, w.s2s_h, q_star);
    }
}

extern "C" void kernel_launch(void* const* d_in, const int* in_sizes, int n_in,
                              void* d_out, int out_size, void* d_ws, size_t ws_size,
                              hipStream_t stream) {
    (void)in_sizes; (void)n_in; (void)out_size; (void)ws_size;

    const float* don_nf = (const float*)d_in[0];
    const float* don_ef = (const float*)d_in[1];
    const float* acc_nf = (const float*)d_in[2];
    const float* acc_ef = (const float*)d_in[3];
    const int* don_src = (const int*)d_in[4];
    const int* don_dst = (const int*)d_in[5];
    const int* acc_src = (const int*)d_in[6];
    const int* acc_dst = (const int*)d_in[7];
    // d_in[8] = node_gid (nodes are contiguous per graph; implicit)

    const int PBASE = 9;
    const int TOWER_N = 26;  // 3 layers * 6 + 2 gamma + 2 beta + 4 s2s
    void* const* Pdon = d_in + PBASE;
    void* const* Pacc = d_in + PBASE + TOWER_N;
    const float* fc1_w = (const float*)d_in[PBASE + 2 * TOWER_N + 0];
    const float* fc1_b = (const float*)d_in[PBASE + 2 * TOWER_N + 1];
    const float* fc2_w = (const float*)d_in[PBASE + 2 * TOWER_N + 2];
    const float* fc2_b = (const float*)d_in[PBASE + 2 * TOWER_N + 3];
    const float* fc3_w = (const float*)d_in[PBASE + 2 * TOWER_N + 4];
    const float* fc3_b = (const float*)d_in[PBASE + 2 * TOWER_N + 5];

    // carve workspace
    char* base = (char*)d_ws;
    size_t off = 0;
    auto carve = [&](size_t bytes) {
        float* p = (float*)(base + off);
        off += (bytes + 255) & ~(size_t)255;
        return p;
    };
    WS w;
    w.Pni    = carve((size_t)NNODES * 512 * 4);
    w.Pnj    = carve((size_t)NNODES * 512 * 4);
    w.Pnode  = carve((size_t)NNODES * 512 * 4);
    w.e_att  = carve((size_t)NEDGES * HEADS * 4);
    w.node_m = carve((size_t)NNODES * HEADS * 4);
    w.node_s = carve((size_t)NNODES * HEADS * 4);
    w.h_out  = carve((size_t)NNODES * 512 * 4);
    w.nfA    = carve((size_t)NNODES * 128 * 4);
    w.nfB    = carve((size_t)NNODES * 128 * 4);
    w.efA    = carve((size_t)NEDGES * 128 * 4);
    w.efB    = carve((size_t)NEDGES * 128 * 4);
    w.bn_mu  = carve(128 * 4);
    w.bn_var = carve(128 * 4);
    w.s2s_h  = carve(BATCH * 64 * 4);
    w.s2s_c  = carve(BATCH * 64 * 4);
    w.q_don  = carve(BATCH * 128 * 4);
    w.q_acc  = carve(BATCH * 128 * 4);

    run_tower(don_nf, don_ef, don_src, don_dst, Pdon, w, w.q_don, stream);
    run_tower(acc_nf, acc_ef, acc_src, acc_dst, Pacc, w, w.q_acc, stream);

    k_head_mlp<<<1, 256, 0, stream>>>(w.q_don, w.q_acc, fc1_w, fc1_b, fc2_w, fc2_b,
                                      fc3_w, fc3_b, (float*)d_out);
}